// TCenterAttentionLayer_7146825580707
// MI455X (gfx1250) — compile-verified
//
#include <hip/hip_runtime.h>
#include <math.h>

// ---------------------------------------------------------------------------
// TCenterAttention fused pipeline for MI455X (gfx1250, wave32, WMMA).
// bs=16, C=256, N=16384, M=64, h=4, d=64. fp32 throughout (memory-bound:
// ~103 GFLOP vs ~1.3 GB HBM traffic at 23.3 TB/s -> keep full precision,
// V_WMMA_F32_16X16X4_F32 for all GEMMs).
// LDS tiles are stored TRANSPOSED ([point][channel], stride 260) so each WMMA
// B fragment {B[k][n],B[k+1][n]} is one aligned, bank-conflict-free
// ds_load_b64, and C tiles store as contiguous runs (b128-able).
// ---------------------------------------------------------------------------

#define BS     16
#define CCH    256
#define NPTS   16384
#define MC     64
#define NT     64               // points per tile
#define NTILES (NPTS / NT)      // 256
#define NTHR   512              // 16 wave32s per block
#define LDP    (CCH + 4)        // 260: even (8B align) and 260%64=4 (no bank conflicts)

typedef float v2f __attribute__((ext_vector_type(2)));
typedef float v8f __attribute__((ext_vector_type(8)));

// fp32 WMMA: D(16x16,f32) = A(16x4,f32) * B(4x16,f32) + C
__device__ __forceinline__ v8f wmma4(v2f a, v2f b, v8f c) {
  return __builtin_amdgcn_wmma_f32_16x16x4_f32(
      /*neg_a=*/false, a, /*neg_b=*/false, b,
      /*c_mod=*/(short)0, c, /*reuse_a=*/false, /*reuse_b=*/false);
}

// ---------------------------------------------------------------------------
// Setup kernels (tiny)
// ---------------------------------------------------------------------------

// G[(h,m), c] = sum_d centers[h*64+d, m] * q_w[h*64+d, c]   (256x256)
__global__ void k_build_G(const float* __restrict__ centers,
                          const float* __restrict__ qw,
                          float* __restrict__ G) {
  int row = blockIdx.x;             // (h,m)
  int h = row >> 6, m = row & 63;
  int c = threadIdx.x;
  float s = 0.f;
  #pragma unroll 4
  for (int dd = 0; dd < 64; ++dd) {
    int ch = h * 64 + dd;
    s += centers[ch * MC + m] * qw[ch * CCH + c];
  }
  G[row * CCH + c] = s;
}

// xv[o, m] = v_b[o] + sum_c v_w[o,c] * centers[c,m]   (256x64)
__global__ void k_build_xv(const float* __restrict__ centers,
                           const float* __restrict__ vw,
                           const float* __restrict__ vb,
                           float* __restrict__ xv) {
  int o = blockIdx.x;
  int m = threadIdx.x;
  float s = vb[o];
  #pragma unroll 4
  for (int c = 0; c < CCH; ++c) s += vw[o * CCH + c] * centers[c * MC + m];
  xv[o * MC + m] = s;
}

// Folded BN: y_bn = rs[o]*gemm + rb[o]
__global__ void k_build_bn(const float* __restrict__ tb,
                           const float* __restrict__ gamma,
                           const float* __restrict__ beta,
                           const float* __restrict__ mean,
                           const float* __restrict__ var,
                           float* __restrict__ rs, float* __restrict__ rb) {
  int o = threadIdx.x;
  float inv = rsqrtf(var[o] + 1e-5f);
  float sc = gamma[o] * inv;
  rs[o] = sc;
  rb[o] = sc * (tb[o] - mean[o]) + beta[o];
}

// Et[b,row] = sum_c G[row,c] * t_embed[b,c]  (energy contribution of t_embed,
// lets the tile loader copy raw local_x with no elementwise add)
__global__ void k_build_Et(const float* __restrict__ G,
                           const float* __restrict__ temb,
                           float* __restrict__ Et) {
  int b = blockIdx.x, row = threadIdx.x;
  __shared__ float ts[CCH];
  ts[row] = temb[b * CCH + row];
  __syncthreads();
  float s = 0.f;
  #pragma unroll 4
  for (int c = 0; c < CCH; ++c) s += G[row * CCH + c] * ts[c];
  Et[b * CCH + row] = s;
}

// ---------------------------------------------------------------------------
// Shared tile stage: transpose-load raw lx tile; E = G@lx + Et ; softmax over
// the 64 m-rows of each head.  Leaves A (softmaxed attention) in est[][],
// transposed: est[j][(h,m)].
// ---------------------------------------------------------------------------
__device__ __forceinline__ void tile_attention(
    const float* __restrict__ lx, const float* __restrict__ G,
    const float* __restrict__ Et, int b, int n0, int tid,
    float (*xt)[LDP], float (*est)[LDP], float* ets) {
  const float* lxb = lx + (size_t)b * CCH * NPTS + n0;
  if (tid < CCH) ets[tid] = Et[b * CCH + tid];

  // -- transpose-load raw lx tile: coalesced float4 global reads
  #pragma unroll
  for (int it = 0; it < 8; ++it) {
    int chunk = tid + it * NTHR;            // 0..4095
    int c = chunk >> 4, j0 = (chunk & 15) << 2;
    float4 v = *(const float4*)(lxb + (size_t)c * NPTS + j0);
    xt[j0 + 0][c] = v.x;
    xt[j0 + 1][c] = v.y;
    xt[j0 + 2][c] = v.z;
    xt[j0 + 3][c] = v.w;
  }
  __syncthreads();

  // -- E = G @ lx (+Et) : wave w -> rows [16w,16w+16) x 64 cols, K=256 step 4
  int wave = tid >> 5, lane = tid & 31, r = lane & 15, kh = lane >> 4;
  int rowbase = wave * 16;
  v8f acc[4] = {};
  for (int k0 = 0; k0 < CCH; k0 += 4) {
    int kr0 = k0 + 2 * kh;                  // K split across lane halves
    v2f a = *(const v2f*)(G + (rowbase + r) * CCH + kr0);
    #pragma unroll
    for (int cs = 0; cs < 4; ++cs) {
      v2f bf = *(const v2f*)(&xt[cs * 16 + r][kr0]);   // one ds_load_b64
      acc[cs] = wmma4(a, bf, acc[cs]);
    }
  }
  int rowSt = rowbase + (kh ? 8 : 0);
  #pragma unroll
  for (int cs = 0; cs < 4; ++cs) {
    int col = cs * 16 + r;
    #pragma unroll
    for (int i = 0; i < 8; ++i)
      est[col][rowSt + i] = acc[cs][i] + ets[rowSt + i];  // contiguous run
  }
  __syncthreads();

  // -- softmax over m within each head: contiguous 64-float segment
  if (tid < 256) {
    int hh = tid >> 6, j = tid & 63;
    float* p = &est[j][hh * 64];
    float mx = -3.4e38f;
    #pragma unroll 8
    for (int m = 0; m < MC; ++m) mx = fmaxf(mx, p[m]);
    float sum = 0.f;
    #pragma unroll 4
    for (int m = 0; m < MC; ++m) {
      float e = __expf(p[m] - mx);
      p[m] = e;
      sum += e;
    }
    float inv = 1.0f / sum;
    #pragma unroll 8
    for (int m = 0; m < MC; ++m) p[m] *= inv;
  }
  __syncthreads();
}

// ---------------------------------------------------------------------------
// Pass 1: deterministic partial column-sums part[b][tile][row] = sum_j A
// ---------------------------------------------------------------------------
__global__ __launch_bounds__(NTHR) void k_colsum(
    const float* __restrict__ lx, const float* __restrict__ G,
    const float* __restrict__ Et, float* __restrict__ part) {
  __shared__ float xt[NT][LDP];
  __shared__ float est[NT][LDP];
  __shared__ float ets[CCH];
  int b = blockIdx.y, n0 = blockIdx.x * NT, tid = threadIdx.x;
  tile_attention(lx, G, Et, b, n0, tid, xt, est, ets);
  if (tid < 256) {
    float s = 0.f;
    #pragma unroll 8
    for (int j = 0; j < NT; ++j) s += est[j][tid];   // conflict-free: stride%64=4
    part[((size_t)b * NTILES + blockIdx.x) * CCH + tid] = s;
  }
}

// ---------------------------------------------------------------------------
// Reduce partials -> S[b,(h,m)]; Vs[b,(h,d),m] = xv/(1e-9+S)  (renorm folded)
// ---------------------------------------------------------------------------
__global__ void k_scaleV(const float* __restrict__ part,
                         const float* __restrict__ xv,
                         float* __restrict__ Vs) {
  int b = blockIdx.x, tid = threadIdx.x;
  __shared__ float S[CCH];
  const float* pb = part + (size_t)b * NTILES * CCH;
  float s = 0.f;
  for (int t = 0; t < NTILES; ++t) s += pb[t * CCH + tid];
  S[tid] = s;
  __syncthreads();
  float* Vsb = Vs + (size_t)b * CCH * MC;
  for (int idx = tid; idx < CCH * MC; idx += 256) {
    int vr = idx >> 6, m = idx & 63, hh = vr >> 6;
    Vsb[idx] = xv[idx] / (1e-9f + S[hh * 64 + m]);
  }
}

// ---------------------------------------------------------------------------
// Pass 2 (fused): recompute A tile; x_r = blkdiag(Vs[b])@A; delta = x_r - x;
// y = t_w@delta; BN+ReLU; out = x + relu(y)
// ---------------------------------------------------------------------------
__global__ __launch_bounds__(NTHR) void k_fused(
    const float* __restrict__ lx, const float* __restrict__ temb,
    const float* __restrict__ G, const float* __restrict__ Et,
    const float* __restrict__ Vs, const float* __restrict__ tw,
    const float* __restrict__ rs_, const float* __restrict__ rb_,
    float* __restrict__ out) {
  __shared__ float xt[NT][LDP];
  __shared__ float est[NT][LDP];
  __shared__ float ets[CCH];
  __shared__ float tes[CCH];
  __shared__ float rss[CCH];
  __shared__ float rbs[CCH];
  int b = blockIdx.y, n0 = blockIdx.x * NT, tid = threadIdx.x;
  if (tid < CCH) {
    tes[tid] = temb[b * CCH + tid];
    rss[tid] = rs_[tid];
    rbs[tid] = rb_[tid];
  }
  tile_attention(lx, G, Et, b, n0, tid, xt, est, ets);

  int wave = tid >> 5, lane = tid & 31, r = lane & 15, kh = lane >> 4;

  // -- x_r: per-head 64x64 GEMM over K=m.  wave w -> head w>>2, strip w&3
  int hh = wave >> 2;
  int rowbase2 = hh * 64 + (wave & 3) * 16;
  const float* Vsb = Vs + (size_t)b * CCH * MC;
  v8f accr[4] = {};
  for (int k0 = 0; k0 < MC; k0 += 4) {
    int kr0 = k0 + 2 * kh;
    v2f a = *(const v2f*)(Vsb + (rowbase2 + r) * MC + kr0);
    #pragma unroll
    for (int cs = 0; cs < 4; ++cs) {
      v2f bf = *(const v2f*)(&est[cs * 16 + r][hh * 64 + kr0]);
      accr[cs] = wmma4(a, bf, accr[cs]);
    }
  }
  __syncthreads();   // all waves done reading A before overwriting est

  // -- delta = x_r - (lx + t_embed) overwrites est (contiguous runs)
  int rowSt2 = rowbase2 + (kh ? 8 : 0);
  #pragma unroll
  for (int cs = 0; cs < 4; ++cs) {
    int col = cs * 16 + r;
    #pragma unroll
    for (int i = 0; i < 8; ++i) {
      int row = rowSt2 + i;
      est[col][row] = accr[cs][i] - (xt[col][row] + tes[row]);
    }
  }
  __syncthreads();

  // -- y = t_w @ delta : wave w -> rows [16w,16w+16), K=256
  int rowbase = wave * 16;
  v8f accy[4] = {};
  for (int k0 = 0; k0 < CCH; k0 += 4) {
    int kr0 = k0 + 2 * kh;
    v2f a = *(const v2f*)(tw + (rowbase + r) * CCH + kr0);
    #pragma unroll
    for (int cs = 0; cs < 4; ++cs) {
      v2f bf = *(const v2f*)(&est[cs * 16 + r][kr0]);
      accy[cs] = wmma4(a, bf, accy[cs]);
    }
  }

  // -- BN + ReLU + residual, store
  float* ob = out + (size_t)b * CCH * NPTS + n0;
  int rowSt = rowbase + (kh ? 8 : 0);
  #pragma unroll
  for (int cs = 0; cs < 4; ++cs) {
    int col = cs * 16 + r;
    #pragma unroll
    for (int i = 0; i < 8; ++i) {
      int row = rowSt + i;
      float xval = xt[col][row] + tes[row];
      float y = rss[row] * accy[cs][i] + rbs[row];
      ob[(size_t)row * NPTS + col] = xval + fmaxf(y, 0.f);
    }
  }
}

// ---------------------------------------------------------------------------
// Launch
// ---------------------------------------------------------------------------
extern "C" void kernel_launch(void* const* d_in, const int* in_sizes, int n_in,
                              void* d_out, int out_size, void* d_ws, size_t ws_size,
                              hipStream_t stream) {
  const float* local_x  = (const float*)d_in[0];
  const float* t_embed  = (const float*)d_in[1];
  /* f_p (d_in[2]) unused: local_channels == 0 */
  const float* centers  = (const float*)d_in[3];
  const float* q_w      = (const float*)d_in[4];
  const float* v_w      = (const float*)d_in[5];
  const float* v_b      = (const float*)d_in[6];
  const float* t_w      = (const float*)d_in[7];
  const float* t_b      = (const float*)d_in[8];
  const float* bn_gamma = (const float*)d_in[9];
  const float* bn_beta  = (const float*)d_in[10];
  const float* bn_mean  = (const float*)d_in[11];
  const float* bn_var   = (const float*)d_in[12];
  float* out = (float*)d_out;

  // workspace layout (floats)
  float* W    = (float*)d_ws;
  float* G    = W;                       // 256*256          = 65536
  float* xv   = W + 65536;               // 256*64           = 16384
  float* Vs   = W + 81920;               // 16*256*64        = 262144
  float* part = W + 344064;              // 16*256*256       = 1048576
  float* rs   = W + 1392640;             // 256
  float* rb   = W + 1392896;             // 256
  float* Et   = W + 1393152;             // 16*256           = 4096
  (void)in_sizes; (void)n_in; (void)out_size; (void)ws_size;

  k_build_G <<<dim3(CCH), dim3(CCH), 0, stream>>>(centers, q_w, G);
  k_build_xv<<<dim3(CCH), dim3(MC),  0, stream>>>(centers, v_w, v_b, xv);
  k_build_bn<<<dim3(1),   dim3(CCH), 0, stream>>>(t_b, bn_gamma, bn_beta,
                                                  bn_mean, bn_var, rs, rb);
  k_build_Et<<<dim3(BS),  dim3(CCH), 0, stream>>>(G, t_embed, Et);
  k_colsum  <<<dim3(NTILES, BS), dim3(NTHR), 0, stream>>>(local_x, G, Et, part);
  k_scaleV  <<<dim3(BS), dim3(CCH), 0, stream>>>(part, xv, Vs);
  k_fused   <<<dim3(NTILES, BS), dim3(NTHR), 0, stream>>>(local_x, t_embed, G, Et,
                                                          Vs, t_w, rs, rb, out);
}